// OptimizedQuantumNoise_42374147342478
// MI455X (gfx1250) — compile-verified
//
#include <hip/hip_runtime.h>
#include <hip/hip_bf16.h>

// ---------------------------------------------------------------------------
// 6-qubit circuit == out[b] = S * | M * psi(z_b) |^2
//   M  = D1 * R1 * D0   (fixed 64x64 complex, from weights only)
//   psi = kron of 6 per-qubit 2-vectors (encoding + layer-0 RY)
//   S  = 6x64 {+1,-1} sign matrix (Z expectations)
// GEMM (64 x 64) x (64 x B) done with V_WMMA_F32_16X16X4_F32.
// ---------------------------------------------------------------------------

typedef __attribute__((ext_vector_type(2))) float v2f;
typedef __attribute__((ext_vector_type(8))) float v8f;

#define WS_MR   0      // MR  [64*64]
#define WS_MI   4096   // MI  [64*64]
#define WS_MIN  8192   // -MI [64*64]
#define WS_C0   12288  // cos(w0j/2) [6]
#define WS_S0   12294  // sin(w0j/2) [6]

struct c32 { float r, i; };
__device__ __forceinline__ c32 cmul(c32 a, c32 b) {
    return { a.r * b.r - a.i * b.i, a.r * b.i + a.i * b.r };
}

// ---------------------------------------------------------------------------
// Setup: build M = D1 * R1 * D0 from weights (w: [2][11] row-major).
// qubit i <-> bit (5-i) of the flat state index.
// ---------------------------------------------------------------------------
__global__ void qsim_setup_kernel(const float* __restrict__ w,
                                  float* __restrict__ ws) {
    __shared__ float d0r[64], d0i[64];
    const int k = threadIdx.x;  // 0..63 = output row
    // ZZ diagonal phases: e^{-i t/2} if bits equal else e^{+i t/2}
    float phi0 = 0.f, phi1 = 0.f;
#pragma unroll
    for (int i = 0; i < 5; ++i) {
        const int bi = (k >> (5 - i)) & 1;
        const int bj = (k >> (4 - i)) & 1;
        const float sgn = (bi == bj) ? -0.5f : 0.5f;
        phi0 += sgn * w[6 + i];   // layer 0 ZZ angles
        phi1 += sgn * w[17 + i];  // layer 1 ZZ angles
    }
    float s, c;
    sincosf(phi0, &s, &c); d0r[k] = c; d0i[k] = s;
    float d1r, d1i; sincosf(phi1, &d1i, &d1r);
    // layer-1 RY half-angle coefficients
    float c1[6], s1[6];
#pragma unroll
    for (int i = 0; i < 6; ++i) sincosf(0.5f * w[11 + i], &s1[i], &c1[i]);
    __syncthreads();
    for (int kp = 0; kp < 64; ++kp) {
        // R1[k,kp] = prod_i RY(w1i)[kb][pb], RY = [[c,-s],[s,c]]
        float re = 1.f;
#pragma unroll
        for (int i = 0; i < 6; ++i) {
            const int kb = (k >> (5 - i)) & 1;
            const int pb = (kp >> (5 - i)) & 1;
            re *= (kb == pb) ? c1[i] : (kb ? s1[i] : -s1[i]);
        }
        const float mr = re * (d1r * d0r[kp] - d1i * d0i[kp]);
        const float mi = re * (d1r * d0i[kp] + d1i * d0r[kp]);
        ws[WS_MR  + k * 64 + kp] = mr;
        ws[WS_MI  + k * 64 + kp] = mi;
        ws[WS_MIN + k * 64 + kp] = -mi;
    }
    if (k < 6) {
        float ss, cc; sincosf(0.5f * w[k], &ss, &cc);
        ws[WS_C0 + k] = cc;
        ws[WS_S0 + k] = ss;
    }
}

// ---------------------------------------------------------------------------
// Main: 128 threads = 4 waves; each wave handles 16 batch columns.
// ---------------------------------------------------------------------------
__global__ __launch_bounds__(128, 1)
void qsim_main_kernel(const float* __restrict__ z,
                      const float* __restrict__ ws,
                      float* __restrict__ out) {
    // per-wave psi staging: 64 rows x 16 cols, row stride 17 (bank-friendly)
    __shared__ float s_psi[4][2][64 * 17];

    const int tid  = threadIdx.x;
    const int wave = tid >> 5;
    const int lane = tid & 31;
    const int n    = lane & 15;   // batch column within tile / matrix row in tile
    const int kh   = lane >> 4;   // lane half
    const int base = blockIdx.x * 64 + wave * 16;
    const int b    = base + n;

    // ---- Phase 1: build psi_b = kron_{j=0..5} u_j, stage to LDS ----------
    const float z0 = z[2 * b + 0];
    const float z1 = z[2 * b + 1];
    float sa, ca, sb, cb;
    sincosf(0.5f * asinf(z0), &sa, &ca);
    sincosf(0.5f * asinf(z1), &sb, &cb);
    const c32 x0{ cb * ca, -sb * ca };   // e^{-ib/2} cos(a/2)
    const c32 x1{ cb * sa,  sb * sa };   // e^{+ib/2} sin(a/2)

    c32 u[6][2];
#pragma unroll
    for (int j = 0; j < 6; ++j) {
        const float c0 = ws[WS_C0 + j], s0 = ws[WS_S0 + j];
        u[j][0] = { c0 * x0.r - s0 * x1.r, c0 * x0.i - s0 * x1.i };
        u[j][1] = { s0 * x0.r + c0 * x1.r, s0 * x0.i + c0 * x1.i };
    }
    c32 t01[4], t012[8], t34[4], t345[8];
#pragma unroll
    for (int p = 0; p < 4; ++p) t01[p]  = cmul(u[0][p >> 1], u[1][p & 1]);
#pragma unroll
    for (int p = 0; p < 8; ++p) t012[p] = cmul(t01[p >> 1], u[2][p & 1]);
#pragma unroll
    for (int p = 0; p < 4; ++p) t34[p]  = cmul(u[3][p >> 1], u[4][p & 1]);
#pragma unroll
    for (int p = 0; p < 8; ++p) t345[p] = cmul(t34[p >> 1], u[5][p & 1]);

    float* psir = &s_psi[wave][0][0];
    float* psii = &s_psi[wave][1][0];
#pragma unroll
    for (int pp = 0; pp < 4; ++pp) {
        const c32 tp = t012[4 * kh + pp];
#pragma unroll
        for (int q = 0; q < 8; ++q) {
            const c32 vq = cmul(tp, t345[q]);
            const int kidx = 8 * (4 * kh + pp) + q;
            psir[kidx * 17 + n] = vq.r;
            psii[kidx * 17 + n] = vq.i;
        }
    }
    __syncthreads();

    // ---- Phase 2: Y = M * Psi via WMMA f32 16x16x4; probs; signed sums ---
    const float* MR = ws + WS_MR;
    const float* MI = ws + WS_MI;
    const float* MN = ws + WS_MIN;

    float acc0 = 0.f, acc1 = 0.f, acc2 = 0.f, acc3 = 0.f, acc4 = 0.f, acc5 = 0.f;

#pragma unroll
    for (int r = 0; r < 4; ++r) {
        v8f yr = {};
        v8f yi = {};
        const float* rowR = MR + (16 * r + n) * 64 + 2 * kh;
        const float* rowI = MI + (16 * r + n) * 64 + 2 * kh;
        const float* rowN = MN + (16 * r + n) * 64 + 2 * kh;
        for (int kk = 0; kk < 16; ++kk) {
            // A fragment: lane (m=n, kh) holds A[m][2kh], A[m][2kh+1]
            const v2f ar = *(const v2f*)(rowR + 4 * kk);
            const v2f ai = *(const v2f*)(rowI + 4 * kk);
            const v2f an = *(const v2f*)(rowN + 4 * kk);
            // B fragment: lane (n, kh) holds B[2kh][n], B[2kh+1][n]
            const int k0 = 4 * kk + 2 * kh;
            const float* pr = psir + k0 * 17 + n;
            const float* pi = psii + k0 * 17 + n;
            const v2f br = { pr[0], pr[17] };
            const v2f bi = { pi[0], pi[17] };
            // Yr += MR*psir + (-MI)*psii ; Yi += MR*psii + MI*psir
            yr = __builtin_amdgcn_wmma_f32_16x16x4_f32(false, ar, false, br, (short)0, yr, false, false);
            yr = __builtin_amdgcn_wmma_f32_16x16x4_f32(false, an, false, bi, (short)0, yr, false, false);
            yi = __builtin_amdgcn_wmma_f32_16x16x4_f32(false, ar, false, bi, (short)0, yi, false, false);
            yi = __builtin_amdgcn_wmma_f32_16x16x4_f32(false, ai, false, br, (short)0, yi, false, false);
        }
        // C/D layout: VGPR v, this lane half -> state row k = 16r + v + 8*kh
        // qubit0<-bit5=r>>1, qubit1<-bit4=r&1, qubit2<-bit3=kh,
        // qubit3<-v&4, qubit4<-v&2, qubit5<-v&1
#pragma unroll
        for (int v = 0; v < 8; ++v) {
            const float p = yr[v] * yr[v] + yi[v] * yi[v];
            acc0 = (r & 2) ? acc0 - p : acc0 + p;
            acc1 = (r & 1) ? acc1 - p : acc1 + p;
            acc2 += p;                          // qubit-2 sign applied below
            acc3 = (v & 4) ? acc3 - p : acc3 + p;
            acc4 = (v & 2) ? acc4 - p : acc4 + p;
            acc5 = (v & 1) ? acc5 - p : acc5 + p;
        }
    }
    acc2 = kh ? -acc2 : acc2;

    // combine the two lane halves (rows k and k+8 live in partner lanes)
    const float e0 = acc0 + __shfl_xor(acc0, 16, 32);
    const float e1 = acc1 + __shfl_xor(acc1, 16, 32);
    const float e2 = acc2 + __shfl_xor(acc2, 16, 32);
    const float e3 = acc3 + __shfl_xor(acc3, 16, 32);
    const float e4 = acc4 + __shfl_xor(acc4, 16, 32);
    const float e5 = acc5 + __shfl_xor(acc5, 16, 32);

    if (kh == 0) {
        float* o = out + (size_t)b * 6;   // 24B-multiple offset -> 8B aligned
        *(v2f*)(o + 0) = (v2f){ e0, e1 };
        *(v2f*)(o + 2) = (v2f){ e2, e3 };
        *(v2f*)(o + 4) = (v2f){ e4, e5 };
    }
}

// ---------------------------------------------------------------------------
extern "C" void kernel_launch(void* const* d_in, const int* in_sizes, int n_in,
                              void* d_out, int out_size, void* d_ws, size_t ws_size,
                              hipStream_t stream) {
    const float* w  = (const float*)d_in[0];   // (2, 11) float32
    const float* z  = (const float*)d_in[1];   // (B, 2) float32
    float* ws       = (float*)d_ws;
    float* o        = (float*)d_out;           // (B, 6) float32
    const int B     = in_sizes[1] / 2;

    qsim_setup_kernel<<<1, 64, 0, stream>>>(w, ws);
    qsim_main_kernel<<<B / 64, 128, 0, stream>>>(z, ws, o);
}